// SparseGeometricAttention_30709016167134
// MI455X (gfx1250) — compile-verified
//
#include <hip/hip_runtime.h>
#include <math.h>

#define BATCH   2
#define SEQ     3072
#define DM      512
#define NH      8
#define DH      64
#define MAXNBR  32
#define R2      0.25f   // radius^2

typedef __attribute__((ext_vector_type(2))) float v2f;
typedef __attribute__((ext_vector_type(4))) float v4f;
typedef __attribute__((ext_vector_type(8))) float v8f;

// ---------------------------------------------------------------------------
// Kernel 1: per-query top-32 nearest neighbors (squared distance, stable on
// ties like argsort), then radius filter (sorted => radius-valid is a prefix).
// ---------------------------------------------------------------------------
__global__ __launch_bounds__(128) void build_neighbors(
    const float* __restrict__ pos, int* __restrict__ nidx, int* __restrict__ ncnt)
{
    int q = blockIdx.x * blockDim.x + threadIdx.x;
    if (q >= BATCH * SEQ) return;
    int b = q / SEQ, s = q - b * SEQ;
    const float* pb = pos + (size_t)b * SEQ * 3;
    float qx = pb[3 * s + 0], qy = pb[3 * s + 1], qz = pb[3 * s + 2];

    float bd[MAXNBR];
    int   bi[MAXNBR];
    int cnt = 0;
    for (int t = 0; t < SEQ; ++t) {
        float dx = qx - pb[3 * t + 0];
        float dy = qy - pb[3 * t + 1];
        float dz = qz - pb[3 * t + 2];
        float d2 = dx * dx + dy * dy + dz * dz;
        if (cnt < MAXNBR || d2 < bd[MAXNBR - 1]) {
            int p = (cnt < MAXNBR) ? cnt : (MAXNBR - 1);
            while (p > 0 && bd[p - 1] > d2) {   // strict > keeps earliest idx on ties
                bd[p] = bd[p - 1];
                bi[p] = bi[p - 1];
                --p;
            }
            bd[p] = d2;
            bi[p] = t;
            if (cnt < MAXNBR) ++cnt;
        }
    }
    // radius filter: sorted ascending, so valid entries form a prefix
    int valid = 0;
    while (valid < cnt && bd[valid] < R2) ++valid;
    int* np = nidx + (size_t)q * MAXNBR;
    for (int j = 0; j < valid; ++j) np[j] = bi[j];
    ncnt[q] = valid;
}

// ---------------------------------------------------------------------------
// Kernel 2: OUT[M x 512] = A[M x 512] @ W[512 x 512] + bias, fp32 WMMA.
// One wave computes a 16x64 output strip (4 N-tiles): each 16x4 A fragment is
// reused across 4 v_wmma_f32_16x16x4_f32 into 4 accumulators, quadrupling
// arithmetic intensity on the A stream. Grid supplies exactly
// (M/16)*(512/64) waves.
//
// VGPR layouts per CDNA5 ISA 7.12.2:
//   A 16x4 : lane<16 holds {K=k0,k0+1} of row (r0+lane); lane>=16 holds
//            {K=k0+2,k0+3} -> one contiguous float2 load per lane.
//   B 4x16 : v0 = row K=k0 (lanes 0-15) / K=k0+2 (lanes 16-31), v1 = K=+1/+3.
//   C/D    : vgpr r -> row r0+r (lanes 0-15) / r0+8+r (lanes 16-31).
// ---------------------------------------------------------------------------
__global__ __launch_bounds__(256) void gemm512_bias_wmma(
    const float* __restrict__ A, const float* __restrict__ W,
    const float* __restrict__ bias, float* __restrict__ out)
{
    const int NG  = DM / 64;                                    // 8 col groups
    int wid  = blockIdx.x * (blockDim.x >> 5) + (threadIdx.x >> 5);
    int tm   = wid / NG;
    int tg   = wid - tm * NG;
    int lane = threadIdx.x & 31;
    int half = lane >> 4;        // 0: lanes 0-15, 1: lanes 16-31
    int lr   = lane & 15;

    int row  = tm * 16 + lr;     // A row for this lane
    int col0 = tg * 64 + lr;     // first B/C/D column for this lane

    const float* ap = A + (size_t)row * DM + 2 * half;
    const float* bp = W + (size_t)(2 * half) * DM + col0;

    v8f acc0 = {}, acc1 = {}, acc2 = {}, acc3 = {};
#pragma unroll 4
    for (int k0 = 0; k0 < DM; k0 += 4) {
        v2f a = *(const v2f*)(ap + k0);          // {K=k0+2h, K=k0+2h+1} of row
        const float* b0 = bp + (size_t)k0 * DM;  // row K=k0+2h
        const float* b1 = b0 + DM;               // row K=k0+2h+1
        v2f bv0 = { b0[0],  b1[0]  };
        v2f bv1 = { b0[16], b1[16] };
        v2f bv2 = { b0[32], b1[32] };
        v2f bv3 = { b0[48], b1[48] };
        acc0 = __builtin_amdgcn_wmma_f32_16x16x4_f32(false, a, false, bv0, (short)0, acc0, false, false);
        acc1 = __builtin_amdgcn_wmma_f32_16x16x4_f32(false, a, false, bv1, (short)0, acc1, false, false);
        acc2 = __builtin_amdgcn_wmma_f32_16x16x4_f32(false, a, false, bv2, (short)0, acc2, false, false);
        acc3 = __builtin_amdgcn_wmma_f32_16x16x4_f32(false, a, false, bv3, (short)0, acc3, false, false);
    }

    float bb0 = bias[col0 +  0];
    float bb1 = bias[col0 + 16];
    float bb2 = bias[col0 + 32];
    float bb3 = bias[col0 + 48];
    float* op = out + (size_t)(tm * 16 + 8 * half) * DM + col0;
#pragma unroll
    for (int r = 0; r < 8; ++r) {
        op[(size_t)r * DM +  0] = acc0[r] + bb0;
        op[(size_t)r * DM + 16] = acc1[r] + bb1;
        op[(size_t)r * DM + 32] = acc2[r] + bb2;
        op[(size_t)r * DM + 48] = acc3[r] + bb3;
    }
}

// ---------------------------------------------------------------------------
// Kernel 3: sparse attention. One wave per (b,s,h). Lane j scores neighbor j
// (<=32 neighbors), wave-shuffle softmax, probs staged through LDS, then each
// lane accumulates output dims d=lane and d=lane+32 of the 64-wide head.
// ---------------------------------------------------------------------------
__global__ __launch_bounds__(128) void sparse_attn(
    const float* __restrict__ Q, const float* __restrict__ K,
    const float* __restrict__ V, const int* __restrict__ nidx,
    const int* __restrict__ ncnt, float* __restrict__ out)
{
    __shared__ float sp[4][MAXNBR];
    __shared__ int   st[4][MAXNBR];

    int w    = threadIdx.x >> 5;
    int lane = threadIdx.x & 31;
    int wid  = blockIdx.x * 4 + w;          // (b*SEQ + s)*NH + h
    int h    = wid & (NH - 1);
    int q    = wid >> 3;                    // b*SEQ + s
    int b    = q / SEQ;

    int cnt = ncnt[q];
    const float* qp = Q + (size_t)q * DM + h * DH;

    float sc = -INFINITY;
    int   t  = 0;
    if (lane < cnt) {
        t = nidx[(size_t)q * MAXNBR + lane];
        const float* kp = K + (size_t)(b * SEQ + t) * DM + h * DH;
        float d = 0.f;
#pragma unroll
        for (int i = 0; i < DH / 4; ++i) {
            v4f qv = *(const v4f*)(qp + 4 * i);
            v4f kv = *(const v4f*)(kp + 4 * i);
            d += qv.x * kv.x + qv.y * kv.y + qv.z * kv.z + qv.w * kv.w;
        }
        sc = d * 0.125f;                    // / sqrt(64)
    }

    // wave32 softmax over neighbor lanes
    float m = sc;
    for (int o = 16; o > 0; o >>= 1) m = fmaxf(m, __shfl_xor(m, o, 32));
    float e = (lane < cnt) ? __expf(sc - m) : 0.f;
    float ssum = e;
    for (int o = 16; o > 0; o >>= 1) ssum += __shfl_xor(ssum, o, 32);
    float p = (ssum > 0.f) ? (e / ssum) : 0.f;   // nan_to_num for empty rows

    sp[w][lane] = p;
    st[w][lane] = t;
    __syncthreads();

    float o0 = 0.f, o1 = 0.f;
#pragma unroll 4
    for (int j = 0; j < MAXNBR; ++j) {
        float pj = sp[w][j];
        const float* vp = V + (size_t)(b * SEQ + st[w][j]) * DM + h * DH;
        o0 = fmaf(pj, vp[lane],      o0);
        o1 = fmaf(pj, vp[lane + 32], o1);
    }
    float* op = out + (size_t)q * DM + h * DH;
    op[lane]      = o0;
    op[lane + 32] = o1;
}

// ---------------------------------------------------------------------------
// Host-side launcher
// ---------------------------------------------------------------------------
extern "C" void kernel_launch(void* const* d_in, const int* in_sizes, int n_in,
                              void* d_out, int out_size, void* d_ws, size_t ws_size,
                              hipStream_t stream) {
    (void)in_sizes; (void)n_in; (void)out_size; (void)ws_size;
    const float* x   = (const float*)d_in[0];
    const float* pos = (const float*)d_in[1];
    const float* Wq  = (const float*)d_in[2];
    const float* bq  = (const float*)d_in[3];
    const float* Wk  = (const float*)d_in[4];
    const float* bk  = (const float*)d_in[5];
    const float* Wv  = (const float*)d_in[6];
    const float* bv  = (const float*)d_in[7];
    const float* Wo  = (const float*)d_in[8];
    const float* bo  = (const float*)d_in[9];
    float* outp = (float*)d_out;

    char*  ws  = (char*)d_ws;
    size_t mat = (size_t)BATCH * SEQ * DM * sizeof(float);   // 12.58 MB each
    float* Qb  = (float*)(ws + 0 * mat);
    float* Kb  = (float*)(ws + 1 * mat);
    float* Vb  = (float*)(ws + 2 * mat);
    float* Ab  = (float*)(ws + 3 * mat);
    int*   nidx = (int*)(ws + 4 * mat);
    int*   ncnt = nidx + (size_t)BATCH * SEQ * MAXNBR;

    build_neighbors<<<(BATCH * SEQ + 127) / 128, 128, 0, stream>>>(pos, nidx, ncnt);

    // (M/16) * (N/64) waves, 8 waves per 256-thread block
    const int gemmBlocks = (BATCH * SEQ / 16) * (DM / 64) / 8;  // 384
    gemm512_bias_wmma<<<gemmBlocks, 256, 0, stream>>>(x, Wq, bq, Qb);
    gemm512_bias_wmma<<<gemmBlocks, 256, 0, stream>>>(x, Wk, bk, Kb);
    gemm512_bias_wmma<<<gemmBlocks, 256, 0, stream>>>(x, Wv, bv, Vb);

    sparse_attn<<<(BATCH * SEQ * NH) / 4, 128, 0, stream>>>(Qb, Kb, Vb, nidx, ncnt, Ab);

    gemm512_bias_wmma<<<gemmBlocks, 256, 0, stream>>>(Ab, Wo, bo, outp);
}